// apool_52286931861675
// MI455X (gfx1250) — compile-verified
//
#include <hip/hip_runtime.h>

// Top-4-mean pooling over rows of 49 f32 (512x2048 rows).
// Memory-bound: ~209 MB total traffic -> ~9us floor @ 23.3 TB/s.
// CDNA5 path: async global->LDS copies (ASYNCcnt) for coalesced staging,
// then branchless top-4 maintenance from conflict-free LDS reads.

#define RPB      256                 // rows per block == blockDim.x
#define ROW      49                  // floats per row
#define TILE_DW  (RPB * ROW)         // 12544 dwords = 50176 bytes LDS

__global__ __launch_bounds__(RPB) void apool_topk4_mean_kernel(
    const float* __restrict__ in, float* __restrict__ out, int num_rows) {
  __shared__ float tile[TILE_DW];

  const int tid = threadIdx.x;
  const long long row0 = (long long)blockIdx.x * RPB;
  const float* gbase = in + row0 * ROW;   // block region: 50176 B, 16B-aligned

  // ---- Stage the block's 50,176 contiguous bytes into LDS asynchronously.
  // 3136 float4 chunks total: 12 x b128 per thread (lane-consecutive, fully
  // coalesced 128B wavefront requests) + 256 x b32 tail (1 per thread).
  {
    // Generic LDS pointer: addr[31:0] is the wave-relative LDS offset the
    // async unit expects in VDST (hardware adds LDS_BASE).
    const unsigned lds_base = (unsigned)(size_t)(&tile[0]);
    const unsigned long long gaddr = (unsigned long long)(size_t)gbase;
#pragma unroll
    for (int k = 0; k < 12; ++k) {
      const unsigned idx16   = (unsigned)(tid + k * RPB);      // float4 index
      const unsigned lds_off = lds_base + idx16 * 16u;
      const unsigned long long ga = gaddr + (unsigned long long)idx16 * 16ull;
      asm volatile("global_load_async_to_lds_b128 %0, %1, off"
                   :: "v"(lds_off), "v"(ga) : "memory");
    }
    { // tail: dwords 12288..12543
      const unsigned idxd    = 12u * RPB * 4u + (unsigned)tid;
      const unsigned lds_off = lds_base + idxd * 4u;
      const unsigned long long ga = gaddr + (unsigned long long)idxd * 4ull;
      asm volatile("global_load_async_to_lds_b32 %0, %1, off"
                   :: "v"(lds_off), "v"(ga) : "memory");
    }
    // Wait for this wave's async copies, then sync the workgroup.
    asm volatile("s_wait_asynccnt 0" ::: "memory");
  }
  __syncthreads();

  // ---- One thread per row. LDS dword index = t*49 + i: 49 is odd, hence
  // invertible mod 64 banks -> conflict-free across each wave32.
  const float* rowp = &tile[tid * ROW];
  float m0 = -__builtin_inff(), m1 = -__builtin_inff();
  float m2 = -__builtin_inff(), m3 = -__builtin_inff();
#pragma unroll
  for (int i = 0; i < ROW; ++i) {
    float x = rowp[i];
    // Branchless insertion into sorted (m0 >= m1 >= m2 >= m3):
    float a0 = fmaxf(x, m0); x = fminf(x, m0); m0 = a0;
    float a1 = fmaxf(x, m1); x = fminf(x, m1); m1 = a1;
    float a2 = fmaxf(x, m2); x = fminf(x, m2); m2 = a2;
    m3 = fmaxf(x, m3);
  }

  const long long r = row0 + tid;
  if (r < num_rows)
    out[r] = (m0 + m1 + m2 + m3) * 0.25f;   // coalesced b32 store
}

extern "C" void kernel_launch(void* const* d_in, const int* in_sizes, int n_in,
                              void* d_out, int out_size, void* d_ws, size_t ws_size,
                              hipStream_t stream) {
  (void)in_sizes; (void)n_in; (void)d_ws; (void)ws_size;
  const float* in = (const float*)d_in[0];   // (512,2048,7,7) f32, contiguous
  float* out = (float*)d_out;                // 512*2048 f32
  const int num_rows = out_size;             // 1,048,576 (multiple of RPB)
  const int blocks = (num_rows + RPB - 1) / RPB;
  apool_topk4_mean_kernel<<<blocks, RPB, 0, stream>>>(in, out, num_rows);
}